// SpikingNet_80367428043417
// MI455X (gfx1250) — compile-verified
//
#include <hip/hip_runtime.h>
#include <math.h>

typedef __attribute__((ext_vector_type(2))) float v2f;
typedef __attribute__((ext_vector_type(8))) float v8f;

// One wave32 computes a 16(batch) x 64(neuron) tile of cur = X @ W^T + bias,
// then applies the fused LIF update (snnTorch Leaky, reset-by-subtraction):
//   reset = (m_prev - thr > 0) ? thr : 0
//   m     = clip(beta,0,1)*m_prev + cur - reset
//   s     = (m - thr > 0) ? 1 : 0
//
// WMMA f32 16x16x4 operand mapping (ISA 7.12.2):
//   A 16x4 : lane l (l16 = l&15, hi = l>>4) holds A[l16][2*hi + {0,1}]  -> float2 load
//   B 4x16 : lane l holds B[2*hi + {0,1}][l16] = W[n0 + l16][k + 2*hi + {0,1}] -> float2 load
//   C 16x16: lane l, vgpr r holds C[8*hi + r][l16]
__global__ __launch_bounds__(32) void lif_layer_f32wmma(
    const float* __restrict__ X,      // [B x K] row-major activations (binary spikes)
    const float* __restrict__ W,      // [N x K] row-major weights
    const float* __restrict__ bias,   // [N]
    const float* __restrict__ m_prev, // [B x N] previous membrane, or nullptr (zeros)
    const float* __restrict__ betaP,  // scalar (device)
    const float* __restrict__ thrP,   // scalar (device)
    float* __restrict__ s_out,        // [B x N] spikes out
    float* __restrict__ m_out,        // [B x N] membrane out
    int K, int N)
{
    const int lane = threadIdx.x;
    const int hi   = lane >> 4;   // lane half selects K sub-pair
    const int l16  = lane & 15;

    const int n0 = blockIdx.x * 64;
    const float* Ap  = X + (size_t)(blockIdx.y * 16 + l16) * K + 2 * hi;
    const float* Bp0 = W + (size_t)(n0 +  0 + l16) * K + 2 * hi;
    const float* Bp1 = W + (size_t)(n0 + 16 + l16) * K + 2 * hi;
    const float* Bp2 = W + (size_t)(n0 + 32 + l16) * K + 2 * hi;
    const float* Bp3 = W + (size_t)(n0 + 48 + l16) * K + 2 * hi;

    v8f acc0 = {}, acc1 = {}, acc2 = {}, acc3 = {};

    #pragma unroll 2
    for (int k = 0; k < K; k += 4) {
        v2f a  = *(const v2f*)(Ap  + k);
        v2f b0 = *(const v2f*)(Bp0 + k);
        v2f b1 = *(const v2f*)(Bp1 + k);
        v2f b2 = *(const v2f*)(Bp2 + k);
        v2f b3 = *(const v2f*)(Bp3 + k);
        acc0 = __builtin_amdgcn_wmma_f32_16x16x4_f32(false, a, false, b0, (short)0, acc0, false, false);
        acc1 = __builtin_amdgcn_wmma_f32_16x16x4_f32(false, a, false, b1, (short)0, acc1, false, false);
        acc2 = __builtin_amdgcn_wmma_f32_16x16x4_f32(false, a, false, b2, (short)0, acc2, false, false);
        acc3 = __builtin_amdgcn_wmma_f32_16x16x4_f32(false, a, false, b3, (short)0, acc3, false, false);
    }

    const float beta = fminf(fmaxf(betaP[0], 0.0f), 1.0f);
    const float thr  = thrP[0];

    v8f accs[4] = {acc0, acc1, acc2, acc3};
    #pragma unroll
    for (int t = 0; t < 4; ++t) {
        const int   ocol = n0 + t * 16 + l16;
        const float bv   = bias[ocol];
        #pragma unroll
        for (int r = 0; r < 8; ++r) {
            const int    brow = blockIdx.y * 16 + hi * 8 + r;
            const size_t idx  = (size_t)brow * N + ocol;
            const float cur   = accs[t][r] + bv;
            const float mp    = m_prev ? m_prev[idx] : 0.0f;
            const float reset = (mp - thr > 0.0f) ? thr : 0.0f;
            const float mn    = beta * mp + cur - reset;
            m_out[idx] = mn;
            s_out[idx] = (mn - thr > 0.0f) ? 1.0f : 0.0f;
        }
    }
}

extern "C" void kernel_launch(void* const* d_in, const int* in_sizes, int n_in,
                              void* d_out, int out_size, void* d_ws, size_t ws_size,
                              hipStream_t stream) {
    (void)in_sizes; (void)n_in; (void)out_size; (void)d_ws; (void)ws_size;

    const float* spikes = (const float*)d_in[0];
    const float* W1 = (const float*)d_in[1];  const float* b1 = (const float*)d_in[2];
    const float* W2 = (const float*)d_in[3];  const float* b2 = (const float*)d_in[4];
    const float* W3 = (const float*)d_in[5];  const float* b3 = (const float*)d_in[6];
    const float* beta1 = (const float*)d_in[7];  const float* thr1 = (const float*)d_in[8];
    const float* beta2 = (const float*)d_in[9];  const float* thr2 = (const float*)d_in[10];
    const float* beta3 = (const float*)d_in[11]; const float* thr3 = (const float*)d_in[12];

    const int T = 100, B = 128, DIN = 1024, DH = 2048, DOUT = 512;

    // d_out layout: s1,s2,s3,m1,m2,m3 concatenated, each [T,B,d]
    float* out = (float*)d_out;
    const size_t SH = (size_t)T * B * DH;
    const size_t SO = (size_t)T * B * DOUT;
    float* s1 = out;
    float* s2 = s1 + SH;
    float* s3 = s2 + SH;
    float* m1 = s3 + SO;
    float* m2 = m1 + SH;
    float* m3 = m2 + SH;

    const dim3 blk(32, 1, 1);
    const dim3 gH(DH / 64, B / 16);    // 32 x 8 workgroups for hidden layers
    const dim3 gO(DOUT / 64, B / 16);  //  8 x 8 workgroups for output layer

    const size_t stepH = (size_t)B * DH;
    const size_t stepO = (size_t)B * DOUT;
    const size_t stepI = (size_t)B * DIN;

    for (int t = 0; t < T; ++t) {
        const size_t offH = (size_t)t * stepH;
        const size_t offO = (size_t)t * stepO;

        lif_layer_f32wmma<<<gH, blk, 0, stream>>>(
            spikes + (size_t)t * stepI, W1, b1,
            t ? (m1 + offH - stepH) : (const float*)nullptr, beta1, thr1,
            s1 + offH, m1 + offH, DIN, DH);

        lif_layer_f32wmma<<<gH, blk, 0, stream>>>(
            s1 + offH, W2, b2,
            t ? (m2 + offH - stepH) : (const float*)nullptr, beta2, thr2,
            s2 + offH, m2 + offH, DH, DH);

        lif_layer_f32wmma<<<gO, blk, 0, stream>>>(
            s2 + offH, W3, b3,
            t ? (m3 + offO - stepO) : (const float*)nullptr, beta3, thr3,
            s3 + offO, m3 + offO, DH, DOUT);
    }
}